// CurvatureGatedGCN_88356067213584
// MI455X (gfx1250) — compile-verified
//
#include <hip/hip_runtime.h>
#include <hip/hip_bf16.h>
#include <math.h>

// ---------------------------------------------------------------------------
// CurvatureGatedGCN for MI455X (gfx1250), wave32 + V_WMMA_F32_16X16X4_F32.
// Persistent-wave GEMMs: weight (B) fragments live in VGPRs across tiles.
// ---------------------------------------------------------------------------

typedef __attribute__((ext_vector_type(2))) float v2f;
typedef __attribute__((ext_vector_type(8))) float v8f;

#define N_NODES 50000
#define N_EDGES 1000000
#define DIM     64
#define MTILES  (N_NODES / 16)   // 3125
#define ETILES  (N_EDGES / 16)   // 62500

#define GEMM_BLOCKS  512         // per N-column, grid-stride over MTILES
#define EDGE_BLOCKS  12500       // grid-stride over ETILES (~5 tiles/wave)

// ---------------- elementwise helpers ----------------

__global__ void k_fill1(float* __restrict__ p, int n) {
    int i = blockIdx.x * blockDim.x + threadIdx.x;
    if (i < n) p[i] = 1.0f;                       // self-loop weight
}

__global__ void k_gate_deg(const float* __restrict__ curv,
                           const int* __restrict__ col,
                           float* __restrict__ gate,
                           float* __restrict__ deg, int E) {
    int e = blockIdx.x * blockDim.x + threadIdx.x;
    if (e < E) {
        float g = 1.0f / (1.0f + __expf(-curv[e] * 0.2f));   // sigmoid(c/5)
        gate[e] = g;
        atomicAdd(&deg[col[e]], g);               // weighted in-degree at target
    }
}

__global__ void k_dinv(const float* __restrict__ deg, float* __restrict__ dinv, int n) {
    int i = blockIdx.x * blockDim.x + threadIdx.x;
    if (i < n) {
        float d = deg[i];
        dinv[i] = (d > 0.0f) ? rsqrtf(d) : 0.0f;
    }
}

__global__ void k_norm(const float* __restrict__ gate,
                       const float* __restrict__ dinv,
                       const int* __restrict__ row,
                       const int* __restrict__ col,
                       float* __restrict__ norm, int E) {
    int e = blockIdx.x * blockDim.x + threadIdx.x;
    if (e < E) norm[e] = dinv[row[e]] * gate[e] * dinv[col[e]];
}

// out[n,d] = b_homo[d] + dinv[n]^2 * xw[n,d]   (GCN self-loop + bias)
__global__ void k_out_init(const float* __restrict__ xw,
                           const float* __restrict__ dinv,
                           const float* __restrict__ b,
                           float* __restrict__ out, int total) {
    int i = blockIdx.x * blockDim.x + threadIdx.x;
    if (i < total) {
        int n = i >> 6, d = i & 63;
        float di = dinv[n];
        out[i] = b[d] + di * di * xw[i];
    }
}

__global__ void k_relu(float* __restrict__ p, int n) {
    int i = blockIdx.x * blockDim.x + threadIdx.x;
    if (i < n) p[i] = fmaxf(p[i], 0.0f);
}

// ---------------- dense GEMM: Y[N,64] = X[N,64] @ W[64,64] via f32 WMMA -----
// blockIdx.y selects the 16-wide N column; B fragments for that column are
// preloaded into 32 VGPRs and reused across a grid-stride loop over M tiles.
__global__ __launch_bounds__(32) void k_gemm64(const float* __restrict__ X,
                                               const float* __restrict__ W,
                                               float* __restrict__ Y) {
    const int lane = threadIdx.x;      // 0..31, full wave active (EXEC all 1s)
    const int half = lane >> 4;        // 0/1
    const int m    = lane & 15;        // A-row / D-column owned by this lane
    const int n0   = blockIdx.y * 16;

    v2f bf[16];                        // B 4x16 f32 fragments, all K steps
#pragma unroll
    for (int k = 0; k < 16; ++k) {
        const int krow = 4 * k + 2 * half;
        bf[k].x = W[(size_t)(krow    ) * DIM + n0 + m];
        bf[k].y = W[(size_t)(krow + 1) * DIM + n0 + m];
    }

    for (int tile = blockIdx.x; tile < MTILES; tile += GEMM_BLOCKS) {
        const int m0 = tile * 16;
        const float* __restrict__ xrow = X + (size_t)(m0 + m) * DIM + 2 * half;
        v8f acc = {};
#pragma unroll
        for (int k = 0; k < 16; ++k) {
            v2f a;                     // A 16x4 f32: lane m, K = 4k+2*half+{0,1}
            a.x = xrow[4 * k + 0];
            a.y = xrow[4 * k + 1];
            acc = __builtin_amdgcn_wmma_f32_16x16x4_f32(
                false, a, false, bf[k], (short)0, acc, false, false);
        }
#pragma unroll
        for (int v = 0; v < 8; ++v)    // C/D: VGPR v holds row v + 8*half
            Y[(size_t)(m0 + v + 8 * half) * DIM + n0 + m] = acc[v];
    }
}

// ---------------- fused edge kernel ----------------------------------------
// Persistent wave, grid-stride over 16-edge tiles:
//   het : D = |x[col]-x[row]| @ W_het   (f32 WMMA, B resident in 128 VGPRs)
//   msg = (1-gate)*(D + b_het) + norm * xw[row]   -> atomicAdd at out[col]
__global__ __launch_bounds__(32) void k_edge(const float* __restrict__ X,
                                             const float* __restrict__ XW,
                                             const float* __restrict__ Whet,
                                             const float* __restrict__ bhet,
                                             const float* __restrict__ gate,
                                             const float* __restrict__ norm,
                                             const int* __restrict__ row,
                                             const int* __restrict__ col,
                                             float* __restrict__ OUT) {
    const int lane = threadIdx.x;
    const int half = lane >> 4;
    const int m    = lane & 15;

    // Preload ALL B fragments of W_het: 4 N-tiles x 16 K-steps x 2 floats.
    v2f bf[64];
    float bv[4];
#pragma unroll
    for (int nt = 0; nt < 4; ++nt) {
        const int n0 = nt * 16;
        bv[nt] = bhet[n0 + m];
#pragma unroll
        for (int k = 0; k < 16; ++k) {
            const int krow = 4 * k + 2 * half;
            bf[nt * 16 + k].x = Whet[(size_t)(krow    ) * DIM + n0 + m];
            bf[nt * 16 + k].y = Whet[(size_t)(krow + 1) * DIM + n0 + m];
        }
    }

    for (int tile = blockIdx.x; tile < ETILES; tile += EDGE_BLOCKS) {
        const int e0 = tile * 16;

        // Gather A tile: A[m][k] = |x[col[e0+m]][k] - x[row[e0+m]][k]|
        const int r = row[e0 + m];
        const int c = col[e0 + m];
        const float* __restrict__ xr = X + (size_t)r * DIM + 2 * half;
        const float* __restrict__ xc = X + (size_t)c * DIM + 2 * half;
        v2f a[16];
#pragma unroll
        for (int k = 0; k < 16; ++k) {
            const int kk = 4 * k;
            a[k].x = fabsf(xc[kk + 0] - xr[kk + 0]);
            a[k].y = fabsf(xc[kk + 1] - xr[kk + 1]);
        }

        for (int nt = 0; nt < 4; ++nt) {
            const int n0 = nt * 16;
            v8f acc = {};
#pragma unroll
            for (int k = 0; k < 16; ++k)
                acc = __builtin_amdgcn_wmma_f32_16x16x4_f32(
                    false, a[k], false, bf[nt * 16 + k], (short)0, acc,
                    false, false);

            // Scatter: lane owns column n0+m; VGPR v holds edge e0+v+8*half.
#pragma unroll
            for (int v = 0; v < 8; ++v) {
                const int e  = e0 + v + 8 * half;
                const int rr = row[e];
                const int cc = col[e];
                const float g  = 1.0f - gate[e];
                const float nm = norm[e];
                const float val = g * (acc[v] + bv[nt])
                                + nm * XW[(size_t)rr * DIM + n0 + m];
                atomicAdd(OUT + (size_t)cc * DIM + n0 + m, val);
            }
        }
    }
}

// ---------------------------------------------------------------------------

extern "C" void kernel_launch(void* const* d_in, const int* in_sizes, int n_in,
                              void* d_out, int out_size, void* d_ws, size_t ws_size,
                              hipStream_t stream) {
    (void)in_sizes; (void)n_in; (void)out_size; (void)ws_size;

    const float* x    = (const float*)d_in[0];
    const int*   ei   = (const int*)  d_in[1];
    const float* curv = (const float*)d_in[2];
    const float* Wh1  = (const float*)d_in[3];   // W_homo1
    const float* bh1  = (const float*)d_in[4];
    const float* Wh2  = (const float*)d_in[5];   // W_homo2
    const float* bh2  = (const float*)d_in[6];
    const float* Wt1  = (const float*)d_in[7];   // W_het1
    const float* bt1  = (const float*)d_in[8];
    const float* Wt2  = (const float*)d_in[9];   // W_het2
    const float* bt2  = (const float*)d_in[10];

    const int N = N_NODES, E = N_EDGES;
    const int* row = ei;
    const int* col = ei + E;

    // Workspace carve-up (16B aligned).
    char* ws = (char*)d_ws;
    auto take = [&](size_t bytes) {
        char* p = ws;
        ws += (bytes + 15) & ~(size_t)15;
        return p;
    };
    float* gate = (float*)take(sizeof(float) * E);
    float* norm = (float*)take(sizeof(float) * E);
    float* deg  = (float*)take(sizeof(float) * N);
    float* dinv = (float*)take(sizeof(float) * N);
    float* xw   = (float*)take(sizeof(float) * (size_t)N * DIM);
    float* h    = (float*)take(sizeof(float) * (size_t)N * DIM);
    float* out  = (float*)d_out;

    const int T = 256;
    const int ND = N * DIM;

    // ---- normalization precompute ----
    k_fill1   <<<(N + T - 1) / T, T, 0, stream>>>(deg, N);
    k_gate_deg<<<(E + T - 1) / T, T, 0, stream>>>(curv, col, gate, deg, E);
    k_dinv    <<<(N + T - 1) / T, T, 0, stream>>>(deg, dinv, N);
    k_norm    <<<(E + T - 1) / T, T, 0, stream>>>(gate, dinv, row, col, norm, E);

    // ---- layer 1: h = relu(gcn1(x) + het1(x)) ----
    k_gemm64  <<<dim3(GEMM_BLOCKS, DIM / 16), 32, 0, stream>>>(x, Wh1, xw);
    k_out_init<<<(ND + T - 1) / T, T, 0, stream>>>(xw, dinv, bh1, h, ND);
    k_edge    <<<EDGE_BLOCKS, 32, 0, stream>>>(x, xw, Wt1, bt1, gate, norm, row, col, h);
    k_relu    <<<(ND + T - 1) / T, T, 0, stream>>>(h, ND);

    // ---- layer 2: out = gcn2(h) + het2(h) ----
    k_gemm64  <<<dim3(GEMM_BLOCKS, DIM / 16), 32, 0, stream>>>(h, Wh2, xw);
    k_out_init<<<(ND + T - 1) / T, T, 0, stream>>>(xw, dinv, bh2, out, ND);
    k_edge    <<<EDGE_BLOCKS, 32, 0, stream>>>(h, xw, Wt2, bt2, gate, norm, row, col, out);
}